// MambaAdaptiveScanpathGenerator_38835094290553
// MI455X (gfx1250) — compile-verified
//
#include <hip/hip_runtime.h>
#include <hip/hip_bf16.h>

typedef __attribute__((ext_vector_type(2))) float v2f;
typedef __attribute__((ext_vector_type(8))) float v8f;

#define H_IMG   256
#define W_IMG   512
#define PATCH   224
#define FS      14
#define L       (FS * FS)        // 196
#define DM      384
#define DH      96               // DM/4 hidden
#define POOL    16
#define NTHREADS 256

// ---- dynamic LDS layout (float offsets) ----
#define OFF_LOCAL   0                        // local features [196][384] fp32
#define OFF_POOLED  (L * DM)                 // +75264 : pooled [3][196]
#define OFF_FW      (OFF_POOLED + 3 * L)     // +588   : fusion weights [208]
#define OFF_IXF     (OFF_FW + 208)           // +208   : sample x coords [16]
#define OFF_IYF     (OFF_IXF + 16)           // +16    : sample y coords [16]
#define LDS_FLOATS  (OFF_IYF + 16)           // = 76092 floats = 304368 bytes (< 320KB WGP LDS)

__device__ __forceinline__ float gelu_tanh(float x) {
    // jax.nn.gelu default (approximate=True)
    float x3 = x * x * x;
    float t  = tanhf(0.7978845608028654f * (x + 0.044715f * x3));
    return 0.5f * x * (1.0f + t);
}

__global__ void __launch_bounds__(NTHREADS)
focus_fused_kernel(const float* __restrict__ images,     // (B,3,256,512)
                   const float* __restrict__ positions,  // (B,2)
                   const float* __restrict__ gfeat,      // (B,196,384)
                   const float* __restrict__ Wp,         // (3,384)
                   const float* __restrict__ bp,         // (384)
                   const float* __restrict__ w1,         // (384,96)
                   const float* __restrict__ b1,         // (96)
                   const float* __restrict__ w2,         // (96,1)
                   const float* __restrict__ b2,         // (1)
                   float* __restrict__ out)              // (B,196,384)
{
    extern __shared__ __align__(16) float lds[];
    const int b    = blockIdx.x;
    const int tid  = threadIdx.x;
    const int lane = tid & 31;      // wave32
    const int wave = tid >> 5;      // 8 waves / block

    // ---- per-batch uniform scalars (replicated per thread, all SALU-friendly) ----
    const float px = fminf(fmaxf(positions[2 * b + 0], 0.0f), 1.0f);
    const float py = fminf(fmaxf(positions[2 * b + 1], 0.0f), 1.0f);
    const int   xt = (int)(px * (float)W_IMG - (float)PATCH * 0.5f);   // trunc toward zero
    const int   x_start = ((xt % W_IMG) + W_IMG) % W_IMG;              // python-style mod
    const float yclip = fminf(fmaxf(py * (float)H_IMG - (float)PATCH * 0.5f,
                                    0.0f), (float)(H_IMG - PATCH));
    const int   y_start = (int)floorf(yclip);

    // ---- stage 0: affine-grid -> bilinear sampling coords (14 each) ----
    if (tid < FS) {
        float u  = (((float)tid + 0.5f) * 2.0f / (float)FS) - 1.0f;
        float gx = ((float)PATCH / (float)W_IMG) * u +
                   ((2.0f * (float)x_start + (float)PATCH) / (float)W_IMG - 1.0f);
        float gy = ((float)PATCH / (float)H_IMG) * u +
                   ((2.0f * (float)y_start + (float)PATCH) / (float)H_IMG - 1.0f);
        lds[OFF_IXF + tid] = (gx + 1.0f) * ((float)FS * 0.5f) - 0.5f;
        lds[OFF_IYF + tid] = (gy + 1.0f) * ((float)FS * 0.5f) - 0.5f;
    }

    // ---- stage 1: wraparound crop + 16x16 avg pool -> pooled[3][196] in LDS ----
    {
        const int x = tid;                                   // patch column, 0..223 active
        const bool act = (x < PATCH);
        const int  xw  = act ? ((x_start + x) % W_IMG) : 0;
        const float* imgb = images + (size_t)b * 3 * H_IMG * W_IMG;
        for (int band = 0; band < 3 * FS; ++band) {
            const int c = band / FS, i = band % FS;
            float s = 0.0f;
            if (act) {
                const float* rowb = imgb + (size_t)c * H_IMG * W_IMG +
                                    (size_t)(y_start + i * POOL) * W_IMG + xw;
                #pragma unroll
                for (int a = 0; a < POOL; ++a) s += rowb[(size_t)a * W_IMG];
            }
            // reduce 16 columns of one pool window (j = x>>4) via wave32 xor-shuffles
            #pragma unroll
            for (int off = 8; off >= 1; off >>= 1) s += __shfl_xor(s, off, 16);
            if (act && ((x & 15) == 0))
                lds[OFF_POOLED + c * L + i * FS + (x >> 4)] = s * (1.0f / 256.0f);
        }
    }
    __syncthreads();

    // ---- stage 2: local[l][d] = pooled[:,l] . Wp[:,d] + bp[d]  (K=3) ----
    for (int idx = tid; idx < L * DM; idx += NTHREADS) {
        const int l = idx / DM, d = idx - l * DM;
        const float p0 = lds[OFF_POOLED + 0 * L + l];
        const float p1 = lds[OFF_POOLED + 1 * L + l];
        const float p2 = lds[OFF_POOLED + 2 * L + l];
        lds[OFF_LOCAL + idx] =
            fmaf(p0, Wp[d], fmaf(p1, Wp[DM + d], fmaf(p2, Wp[2 * DM + d], bp[d])));
    }
    __syncthreads();

    // ---- stage 3: fw = sigmoid( gelu(local @ w1 + b1) @ w2 + b2 )  via f32 WMMA ----
    // M = 196 (13 tiles of 16), N = 96 (6 tiles of 16), K = 384 (96 steps of 4)
    {
        const int half = lane >> 4;      // 0: lanes 0-15, 1: lanes 16-31
        const int l15  = lane & 15;
        const float b2v = b2[0];
        float w2v[6], b1v[6];
        #pragma unroll
        for (int nt = 0; nt < 6; ++nt) { w2v[nt] = w2[nt * 16 + l15];
                                         b1v[nt] = b1[nt * 16 + l15]; }

        for (int mt = wave; mt < 13; mt += 8) {
            const int mbase = mt * 16;
            // clamp pad rows (196..207) to a valid LDS row; results discarded
            const int arow = (mbase + l15 < L) ? (mbase + l15) : (L - 1);
            const float* lrow = &lds[OFF_LOCAL + arow * DM];

            v8f acc[6];
            const v8f zacc = {};
            #pragma unroll
            for (int nt = 0; nt < 6; ++nt) acc[nt] = zacc;

            for (int k = 0; k < DM; k += 4) {
                // A 16x4 fp32 layout (ISA 7.12.2): v0 = {K0 | K2}, v1 = {K1 | K3}
                const int kk = k + 2 * half;
                v2f afrag;
                afrag.x = lrow[kk];
                afrag.y = lrow[kk + 1];
                #pragma unroll
                for (int nt = 0; nt < 6; ++nt) {
                    const float* wrow = w1 + (size_t)kk * DH + nt * 16 + l15;
                    v2f bfrag;
                    bfrag.x = wrow[0];      // B row K=kk   (N = lane)
                    bfrag.y = wrow[DH];     // B row K=kk+1
                    acc[nt] = __builtin_amdgcn_wmma_f32_16x16x4_f32(
                        false, afrag, false, bfrag, (short)0, acc[nt], false, false);
                }
            }

            // epilogue: hidden -> gelu -> dot w2 (per-lane partial over N) -> reduce
            float rowacc[8];
            #pragma unroll
            for (int r = 0; r < 8; ++r) rowacc[r] = 0.0f;
            #pragma unroll
            for (int nt = 0; nt < 6; ++nt) {
                #pragma unroll
                for (int r = 0; r < 8; ++r) {
                    const float h = acc[nt][r] + b1v[nt];
                    rowacc[r] = fmaf(gelu_tanh(h), w2v[nt], rowacc[r]);
                }
            }
            #pragma unroll
            for (int r = 0; r < 8; ++r) {
                float s = rowacc[r];
                #pragma unroll
                for (int off = 8; off >= 1; off >>= 1) s += __shfl_xor(s, off, 16);
                const int m = mbase + r + 8 * half;   // C layout: vgpr r -> M=r (+8 hi half)
                if (l15 == 0 && m < L)
                    lds[OFF_FW + m] = 1.0f / (1.0f + __expf(-(s + b2v)));
            }
        }
    }
    __syncthreads();

    // ---- stage 4: bilinear grid_sample from LDS + gated residual update ----
    {
        const float* gfb  = gfeat + (size_t)b * L * DM;
        float*       outb = out   + (size_t)b * L * DM;
        for (int idx = tid; idx < L * (DM / 4); idx += NTHREADS) {
            const int l  = idx / (DM / 4);
            const int d4 = idx - l * (DM / 4);
            const int i = l / FS, j = l - i * FS;
            const float ix = lds[OFF_IXF + j];
            const float iy = lds[OFF_IYF + i];
            const float x0f = floorf(ix), y0f = floorf(iy);
            const int   x0 = (int)x0f,   y0 = (int)y0f;
            const float wx1 = ix - x0f, wx0 = 1.0f - wx1;
            const float wy1 = iy - y0f, wy0 = 1.0f - wy1;

            float4 s = make_float4(0.f, 0.f, 0.f, 0.f);
            #pragma unroll
            for (int cy = 0; cy < 2; ++cy) {
                const int yy = y0 + cy;
                if (yy < 0 || yy >= FS) continue;
                const float wy = cy ? wy1 : wy0;
                #pragma unroll
                for (int cx = 0; cx < 2; ++cx) {
                    const int xx = x0 + cx;
                    if (xx < 0 || xx >= FS) continue;
                    const float w = wy * (cx ? wx1 : wx0);
                    const float4 v = *(const float4*)&lds[OFF_LOCAL + (yy * FS + xx) * DM + 4 * d4];
                    s.x = fmaf(w, v.x, s.x); s.y = fmaf(w, v.y, s.y);
                    s.z = fmaf(w, v.z, s.z); s.w = fmaf(w, v.w, s.w);
                }
            }
            const float fwv = lds[OFF_FW + l];
            const float4 g = *(const float4*)&gfb[4 * idx];
            float4 o;
            o.x = fmaf(fwv, s.x, g.x); o.y = fmaf(fwv, s.y, g.y);
            o.z = fmaf(fwv, s.z, g.z); o.w = fmaf(fwv, s.w, g.w);
            *(float4*)&outb[4 * idx] = o;
        }
    }
}

extern "C" void kernel_launch(void* const* d_in, const int* in_sizes, int n_in,
                              void* d_out, int out_size, void* d_ws, size_t ws_size,
                              hipStream_t stream) {
    const float* images    = (const float*)d_in[0];
    const float* positions = (const float*)d_in[1];
    const float* gfeat     = (const float*)d_in[2];
    const float* Wp        = (const float*)d_in[3];
    const float* bp        = (const float*)d_in[4];
    const float* w1        = (const float*)d_in[5];
    const float* b1        = (const float*)d_in[6];
    const float* w2        = (const float*)d_in[7];
    const float* b2        = (const float*)d_in[8];
    // d_in[9] = seq_len (unused)
    float* out = (float*)d_out;

    const int B = in_sizes[1] / 2;                      // positions = (B,2)
    const size_t shbytes = (size_t)LDS_FLOATS * sizeof(float);  // ~304 KB < 320 KB WGP LDS

    focus_fused_kernel<<<B, NTHREADS, shbytes, stream>>>(
        images, positions, gfeat, Wp, bp, w1, b1, w2, b2, out);
}